// NormalizedDistanceLoss_52115133169717
// MI455X (gfx1250) — compile-verified
//
#include <hip/hip_runtime.h>
#include <hip/hip_bf16.h>
#include <math.h>

typedef __attribute__((ext_vector_type(2))) float v2f;
typedef __attribute__((ext_vector_type(8))) float v8f;

#define KDIM 512  // feature dimension (fixed by reference)

// Workspace layout (floats): ws[0]=S accumulator, ws[1]=max(sq) as uint bits,
// ws[2..2+KDIM) = column sums s[k].

__global__ void ndl_init(float* __restrict__ ws) {
    int i = blockIdx.x * blockDim.x + threadIdx.x;
    if (i < 2 + KDIM) ws[i] = 0.0f;  // 0 bits == 0.0f, valid init for max of sq>=0
}

// One wave (32 lanes) owns one 16-row tile. Computes sq_i = sum_k x[i][k]^2 for
// its 16 rows using V_WMMA_F32_16X16X4_F32 with B = ones, then reduces the tile
// to (sum, max) and accumulates into ws[0]/ws[1].
__global__ void ndl_rowpass(const float* __restrict__ x, float* __restrict__ ws,
                            int nrows) {
    const int lane = threadIdx.x & 31;
    const int wave = threadIdx.x >> 5;
    const int tile = blockIdx.x * (blockDim.x >> 5) + wave;
    if (tile * 16 >= nrows) return;  // wave-uniform exit (EXEC all-1s for WMMA)

    // A-matrix 16x4 f32 layout: lanes 0-15 -> row M=lane, K=0(v0),K=1(v1);
    // lanes 16-31 -> row M=lane-16, K=2(v0),K=3(v1).
    const int row  = tile * 16 + (lane & 15);
    const int koff = (lane >> 4) << 1;
    const float* p = x + (size_t)row * KDIM + koff;

    v8f acc = {};
#if __has_builtin(__builtin_amdgcn_wmma_f32_16x16x4_f32)
    const v2f bones = {1.0f, 1.0f};
    #pragma unroll 4
    for (int k = 0; k < KDIM; k += 4) {
        float2 v = *reinterpret_cast<const float2*>(p + k);
        v2f a;
        a.x = v.x * v.x;
        a.y = v.y * v.y;
        // D[m][n] = sum_k A[m][k]*1 + C : every column n holds the row sum.
        acc = __builtin_amdgcn_wmma_f32_16x16x4_f32(
            /*neg_a=*/false, a, /*neg_b=*/false, bones,
            /*c_mod=*/(short)0, acc, /*reuse_a=*/false, /*reuse_b=*/false);
    }
    // C/D layout: VGPR j, lanes 0-15 hold M=j; lanes 16-31 hold M=j+8.
    float lsum = 0.0f, lmax = 0.0f;
    #pragma unroll
    for (int j = 0; j < 8; ++j) {
        lsum += acc[j];
        lmax  = fmaxf(lmax, acc[j]);
    }
    // Halves are redundant across lanes; xor-16 merges rows 0-7 with rows 8-15.
    lsum += __shfl_xor(lsum, 16, 32);
    lmax  = fmaxf(lmax, __shfl_xor(lmax, 16, 32));
#else
    // VALU fallback: per-lane partial sums of squares, then lane reductions.
    float a0 = 0.0f;
    for (int k = 0; k < KDIM; k += 4) {
        float2 v = *reinterpret_cast<const float2*>(p + k);
        a0 += v.x * v.x + v.y * v.y;
    }
    float sq = a0 + __shfl_xor(a0, 16, 32);  // per-row sq in lanes 0..15 (dup 16..31)
    float lsum = sq, lmax = sq;
    for (int off = 1; off < 16; off <<= 1) {
        lsum += __shfl_xor(lsum, off, 32);
        lmax  = fmaxf(lmax, __shfl_xor(lmax, off, 32));
    }
#endif
    if (lane == 0) {
        atomicAdd(ws, lsum);
        atomicMax(reinterpret_cast<unsigned int*>(ws + 1), __float_as_uint(lmax));
    }
}

// Column sums: thread = column (fully coalesced row-major sweep), each block
// owns a contiguous row range, one atomicAdd per (block, column).
__global__ void ndl_colpass(const float* __restrict__ x, float* __restrict__ ws,
                            int rows_per_block) {
    const int c = threadIdx.x;  // 0..KDIM-1
    const float* p = x + (size_t)blockIdx.x * rows_per_block * KDIM + c;
    float acc = 0.0f;
    #pragma unroll 4
    for (int r = 0; r < rows_per_block; ++r) {
        acc += p[(size_t)r * KDIM];
    }
    atomicAdd(ws + 2 + c, acc);
}

// loss = (N*S - ||s||^2) / (sqrt(max sq) * N(N-1)/2)
__global__ void ndl_finalize(const float* __restrict__ ws, float* __restrict__ out,
                             int nrows) {
    __shared__ float red[256];
    const int t = threadIdx.x;
    float a = ws[2 + t];
    float b = ws[2 + 256 + t];
    red[t] = a * a + b * b;
    __syncthreads();
    for (int s = 128; s > 0; s >>= 1) {
        if (t < s) red[t] += red[t + s];
        __syncthreads();
    }
    if (t == 0) {
        double S  = (double)ws[0];
        float  mx = __uint_as_float(*reinterpret_cast<const unsigned int*>(ws + 1));
        double N  = (double)nrows;
        double count = N * (N - 1.0) * 0.5;
        double total = N * S - (double)red[0];
        out[0] = (float)(total / ((double)sqrtf(mx) * count));
    }
}

extern "C" void kernel_launch(void* const* d_in, const int* in_sizes, int n_in,
                              void* d_out, int out_size, void* d_ws, size_t ws_size,
                              hipStream_t stream) {
    const float* x  = (const float*)d_in[0];
    float*       out = (float*)d_out;
    float*       ws  = (float*)d_ws;   // needs (2+KDIM)*4 = 2056 bytes

    const int n = in_sizes[0] / KDIM;  // 8192 rows

    // 1) zero accumulators
    ndl_init<<<(2 + KDIM + 255) / 256, 256, 0, stream>>>(ws);

    // 2) per-row sq via WMMA; one wave per 16-row tile, 8 waves/block
    const int tiles = n / 16;                 // 512
    const int wavesPerBlock = 8;              // 256 threads
    ndl_rowpass<<<(tiles + wavesPerBlock - 1) / wavesPerBlock,
                  32 * wavesPerBlock, 0, stream>>>(x, ws, n);

    // 3) column sums; 64 blocks x 512 threads, coalesced
    const int nblk = 64;
    ndl_colpass<<<nblk, KDIM, 0, stream>>>(x, ws, n / nblk);

    // 4) finalize scalar loss
    ndl_finalize<<<1, 256, 0, stream>>>(ws, out, n);
}